// NTN_74371653697762
// MI455X (gfx1250) — compile-verified
//
#include <hip/hip_runtime.h>
#include <math.h>

typedef __attribute__((ext_vector_type(2))) float v2f;
typedef __attribute__((ext_vector_type(8))) float v8f;

#define DIM       128
#define KSLICES   4
#define NREL      64
#define TILE      16
#define PITCH     132   // 128 + 4 pad -> bank-conflict-free strided row access

// ---------------- binning kernels ----------------

__global__ void ntn_bin_zero(int* __restrict__ counts) {
    if (threadIdx.x < NREL) counts[threadIdx.x] = 0;
}

__global__ void ntn_bin_count(const int* __restrict__ relations,
                              int* __restrict__ counts,
                              int* __restrict__ slot, int B) {
    int b = blockIdx.x * blockDim.x + threadIdx.x;
    if (b < B) {
        int r = relations[b];
        slot[b] = atomicAdd(&counts[r], 1);
    }
}

__global__ void ntn_bin_scan(const int* __restrict__ counts,
                             int* __restrict__ offsets) {
    if (threadIdx.x == 0) {
        int acc = 0;
        for (int r = 0; r < NREL; ++r) { offsets[r] = acc; acc += counts[r]; }
    }
}

__global__ void ntn_bin_scatter(const int* __restrict__ relations,
                                const int* __restrict__ offsets,
                                const int* __restrict__ slot,
                                int* __restrict__ order, int B) {
    int b = blockIdx.x * blockDim.x + threadIdx.x;
    if (b < B) {
        int r = relations[b];
        order[offsets[r] + slot[b]] = b;
    }
}

// ---------------- main NTN kernel: one wave per (relation, 16-row tile) ----------------

__global__ void __launch_bounds__(32)
ntn_main(const int*   __restrict__ heads,
         const int*   __restrict__ tails,
         const float* __restrict__ emb,     // [N_ENTITIES, 128]
         const float* __restrict__ W,       // [64, 4, 128, 128]
         const float* __restrict__ V,       // [64, 4, 256]
         const float* __restrict__ Bp,      // [64, 4]
         const float* __restrict__ U,       // [64, 4]
         const int*   __restrict__ counts,
         const int*   __restrict__ offsets,
         const int*   __restrict__ order,
         float*       __restrict__ out)
{
    const int r = blockIdx.y;
    const int t = blockIdx.x;
    const int cnt = counts[r];
    if (t * TILE >= cnt) return;                 // uniform across wave
    const int base = offsets[r] + t * TILE;
    const int nv   = min(TILE, cnt - t * TILE);
    const int lane = threadIdx.x;

    __shared__ float e1s[TILE * PITCH];
    __shared__ float e2s[TILE * PITCH];
    __shared__ float wlds[TILE * PITCH];
    __shared__ float red[32 * 8];
    __shared__ float gbs[TILE][KSLICES];
    __shared__ int   bidx[TILE];

    if (lane < TILE) {
        int i = (lane < nv) ? lane : (nv - 1);   // pad tail with last valid row
        bidx[lane] = order[base + i];
    }
    __syncthreads();

    // Gather E1/E2 tiles into LDS, coalesced (each lane: one float4 per row)
    for (int rr = 0; rr < TILE; ++rr) {
        int b = bidx[rr];
        const float4* s1 = (const float4*)(emb + (size_t)heads[b] * DIM);
        const float4* s2 = (const float4*)(emb + (size_t)tails[b] * DIM);
        float4 x1 = s1[lane];
        float4 x2 = s2[lane];
        *(float4*)&e1s[rr * PITCH + lane * 4] = x1;
        *(float4*)&e2s[rr * PITCH + lane * 4] = x2;
    }
    __syncthreads();

    // WMMA f32 16x16x4 fragment coordinates:
    //   A (16x4):  lane -> M = lane&15 ; lane-half -> K pair {0,1} / {2,3}
    //   B (4x16):  lane -> N = lane&15 ; lane-half -> K pair {0,1} / {2,3}
    const int m  = lane & 15;
    const int j0 = (lane >> 4) * 2;

    // Preload all 32 A-fragments of the E2 tile (reused for every k and i0)
    v2f afrag[32];
#pragma unroll
    for (int s = 0; s < 32; ++s) {
        const float* p = &e2s[m * PITCH + 4 * s + j0];   // ds_load_b64
        v2f a; a.x = p[0]; a.y = p[1];
        afrag[s] = a;
    }

    const float* Wr = W + (size_t)r * KSLICES * DIM * DIM;

    for (int k = 0; k < KSLICES; ++k) {
        const float* Wk = Wr + (size_t)k * DIM * DIM;
        float part[8];
#pragma unroll
        for (int v = 0; v < 8; ++v) part[v] = 0.f;

        for (int i0 = 0; i0 < DIM; i0 += TILE) {
            // Stage W[k, i0..i0+15, :] (row-major) into LDS, coalesced
            __syncthreads();
            for (int rr = 0; rr < TILE; ++rr) {
                const float4* sw = (const float4*)(Wk + (size_t)(i0 + rr) * DIM);
                *(float4*)&wlds[rr * PITCH + lane * 4] = sw[lane];
            }
            __syncthreads();
            if (i0 + TILE < DIM)   // hint next W tile into cache (global_prefetch_b8)
                __builtin_prefetch(Wk + (size_t)(i0 + TILE) * DIM + lane * 32, 0, 1);

            // Z_tile[16x16] = E2[16x128] * W[k, i0:i0+16, :]^T  (K-dim = 128, 32 steps)
            v8f c = {};
#pragma unroll
            for (int s = 0; s < 32; ++s) {
                const float* q = &wlds[m * PITCH + 4 * s + j0]; // ds_load_b64
                v2f bfrag; bfrag.x = q[0]; bfrag.y = q[1];
                c = __builtin_amdgcn_wmma_f32_16x16x4_f32(
                        /*neg_a=*/false, afrag[s],
                        /*neg_b=*/false, bfrag,
                        /*c_mod=*/(short)0, c,
                        /*reuse_a=*/false, /*reuse_b=*/false);
            }

            // Elementwise-multiply C by E1 and accumulate row partials.
            // C layout: element (M = v + 8*(lane>>4), N = lane&15) in c[v].
#pragma unroll
            for (int v = 0; v < 8; ++v) {
                int mm = v + 8 * (lane >> 4);
                part[v] += c[v] * e1s[mm * PITCH + i0 + m];
            }
        }

        // Cross-lane reduce over N (16 lanes per half) -> g_b[m, k]
#pragma unroll
        for (int v = 0; v < 8; ++v) red[lane * 8 + v] = part[v];
        __syncthreads();
        if (lane < TILE) {
            int hi2 = lane >> 3, v = lane & 7;
            float gb = 0.f;
            for (int n = 0; n < 16; ++n) gb += red[(hi2 * 16 + n) * 8 + v];
            gbs[lane][k] = gb;
        }
        __syncthreads();
    }

    // Linear term, tanh, weighted sum, sigmoid — one lane per batch row
    if (lane < TILE) {
        const int mrow = lane;
        const float* Vr = V + (size_t)r * KSLICES * 2 * DIM;
        float score = 0.f;
#pragma unroll
        for (int k = 0; k < KSLICES; ++k) {
            const float* vk = Vr + (size_t)k * 2 * DIM;
            float ga = 0.f;
            for (int i = 0; i < DIM; ++i) ga += vk[i]       * e1s[mrow * PITCH + i];
            for (int i = 0; i < DIM; ++i) ga += vk[DIM + i] * e2s[mrow * PITCH + i];
            float z = ga + gbs[mrow][k] + Bp[r * KSLICES + k];
            score += U[r * KSLICES + k] * tanhf(z);
        }
        float pred = 1.f / (1.f + expf(-score));
        if (mrow < nv) out[bidx[mrow]] = pred;
    }
}

// ---------------- launch ----------------

extern "C" void kernel_launch(void* const* d_in, const int* in_sizes, int n_in,
                              void* d_out, int out_size, void* d_ws, size_t ws_size,
                              hipStream_t stream) {
    const int*   heads     = (const int*)  d_in[0];
    const int*   tails     = (const int*)  d_in[1];
    const int*   relations = (const int*)  d_in[2];
    const float* emb       = (const float*)d_in[3];
    const float* W         = (const float*)d_in[4];
    const float* V         = (const float*)d_in[5];
    const float* Bp        = (const float*)d_in[6];
    const float* U         = (const float*)d_in[7];
    float*       out       = (float*)d_out;
    const int B = in_sizes[0];   // 4096

    int* counts  = (int*)d_ws;          // [64]
    int* offsets = counts + NREL;       // [64]
    int* slot    = offsets + NREL;      // [B]
    int* order   = slot + B;            // [B]

    ntn_bin_zero   <<<1, 64, 0, stream>>>(counts);
    ntn_bin_count  <<<(B + 255) / 256, 256, 0, stream>>>(relations, counts, slot, B);
    ntn_bin_scan   <<<1, 1, 0, stream>>>(counts, offsets);
    ntn_bin_scatter<<<(B + 255) / 256, 256, 0, stream>>>(relations, offsets, slot, order, B);

    dim3 grid((B + TILE - 1) / TILE, NREL);   // tiles x relations; empty tiles early-exit
    ntn_main<<<grid, 32, 0, stream>>>(heads, tails, emb, W, V, Bp, U,
                                      counts, offsets, order, out);
}